// GraphAttention_9225589752232
// MI455X (gfx1250) — compile-verified
//
#include <hip/hip_runtime.h>
#include <hip/hip_bf16.h>
#include <math.h>

typedef __attribute__((ext_vector_type(2))) float v2f;
typedef __attribute__((ext_vector_type(8))) float v8f;

#define DIMS 128
#define WV   2      // waves per block in the edge-aggregation kernel
#define TILE 16     // edges per wave (one WMMA tile)

// monotone float<->uint mapping so atomicMax(u32) == float max
__device__ __forceinline__ unsigned f2mono(float f) {
  unsigned u = __float_as_uint(f);
  return (u & 0x80000000u) ? ~u : (u | 0x80000000u);
}
__device__ __forceinline__ float mono2f(unsigned u) {
  return (u & 0x80000000u) ? __uint_as_float(u & 0x7fffffffu)
                           : __uint_as_float(~u);
}

// ---------------------------------------------------------------------------
// Kernel 1: feats = tanh(features) -> out[:, :128]; zero accumulator half;
//           init per-row softmax state. One wave per node, float4 per lane.
__global__ void k_nodes_init(const float* __restrict__ features,
                             float* __restrict__ out,
                             unsigned* __restrict__ m_ws,
                             float* __restrict__ ssum, int N) {
  const int wid = threadIdx.x >> 5, lane = threadIdx.x & 31;
  const int n = blockIdx.x * (blockDim.x >> 5) + wid;
  if (n >= N) return;
  float4 f = reinterpret_cast<const float4*>(features + (size_t)n * DIMS)[lane];
  float4 t;
  t.x = tanhf(f.x); t.y = tanhf(f.y); t.z = tanhf(f.z); t.w = tanhf(f.w);
  float* o = out + (size_t)n * (2 * DIMS);
  reinterpret_cast<float4*>(o)[lane] = t;
  float4 z = make_float4(0.f, 0.f, 0.f, 0.f);
  reinterpret_cast<float4*>(o + DIMS)[lane] = z;
  if (lane == 0) { m_ws[n] = f2mono(-INFINITY); ssum[n] = 0.0f; }
}

// ---------------------------------------------------------------------------
// Kernel 2: per relation r: norm[r] = ||rel_emb[r]||, adot[r] = rel_emb[r].k
// One wave per relation, wave32 shuffle reduction.
__global__ void k_rel(const float* __restrict__ rel_emb,
                      const float* __restrict__ attn_kernel,
                      float* __restrict__ norm_ws,
                      float* __restrict__ adot_ws, int R) {
  const int wid = threadIdx.x >> 5, lane = threadIdx.x & 31;
  const int r = blockIdx.x * (blockDim.x >> 5) + wid;
  if (r >= R) return;
  float4 re = reinterpret_cast<const float4*>(rel_emb + (size_t)r * DIMS)[lane];
  float4 ak = reinterpret_cast<const float4*>(attn_kernel)[lane];
  float sq = re.x * re.x + re.y * re.y + re.z * re.z + re.w * re.w;
  float pa = re.x * ak.x + re.y * ak.y + re.z * ak.z + re.w * ak.w;
  for (int off = 16; off > 0; off >>= 1) {
    sq += __shfl_xor(sq, off, 32);
    pa += __shfl_xor(pa, off, 32);
  }
  if (lane == 0) { norm_ws[r] = sqrtf(sq); adot_ws[r] = pa; }
}

// ---------------------------------------------------------------------------
// Kernel 3: per-edge attention logit + per-row max (segment_max).
__global__ void k_edge_logit(const int* __restrict__ adj,
                             const int* __restrict__ ridx,
                             const float* __restrict__ rval,
                             const float* __restrict__ norm_ws,
                             const float* __restrict__ adot_ws,
                             float* __restrict__ logit_ws,
                             unsigned* __restrict__ m_ws, int T) {
  const int t = blockIdx.x * blockDim.x + threadIdx.x;
  if (t >= T) return;
  const int row = adj[t];
  const int rel = ridx[T + t];
  const float rv = rval[t];
  const float s = rv / fmaxf(rv * norm_ws[rel], 1e-12f);  // tri_rel scale
  const float logit = s * adot_ws[rel];
  logit_ws[t] = logit;
  atomicMax(&m_ws[row], f2mono(logit));
}

// ---------------------------------------------------------------------------
// Kernel 4: heavy pass. One wave per 16-edge tile:
//   - stage 16 feature rows + 16 relation rows in LDS (padded rows)
//   - 16 edge dot products = diagonal of 16x16 product, computed with
//     32 chained V_WMMA_F32_16X16X4_F32 on the matrix pipe (K = 128)
//   - scatter e*(f - 2*dot*tri) into out[:,128:] with f32 atomics
__global__ void k_edge_agg(const float* __restrict__ rel_emb,
                           const int* __restrict__ adj,
                           const int* __restrict__ ridx,
                           const float* __restrict__ rval,
                           const float* __restrict__ norm_ws,
                           const float* __restrict__ logit_ws,
                           const unsigned* __restrict__ m_ws,
                           float* __restrict__ ssum,
                           float* __restrict__ out, int T) {
  __shared__ float sA[WV][TILE][DIMS + 4];  // gathered tanh(features[col])
  __shared__ float sB[WV][TILE][DIMS + 4];  // gathered rel_emb[rel] (unscaled)
  __shared__ float sDot[WV][TILE];
  __shared__ float sS[WV][TILE];
  __shared__ float sE[WV][TILE];
  __shared__ int   sRow[WV][TILE];
  __shared__ int   sCol[WV][TILE];
  __shared__ int   sRel[WV][TILE];

  const int w = threadIdx.x >> 5;
  const int lane = threadIdx.x & 31;
  const int t0 = (blockIdx.x * WV + w) * TILE;
  if (t0 >= T) return;  // wave-uniform exit; no block barriers used below

  // --- per-edge scalars (lanes 0..15, one edge each) ---
  if (lane < TILE) {
    const int t = t0 + lane;
    int rowv = 0, colv = 0, relv = 0;
    float sc = 0.0f, ev = 0.0f;
    if (t < T) {
      rowv = adj[t];
      colv = adj[T + t];
      relv = ridx[T + t];
      const float rv = rval[t];
      sc = rv / fmaxf(rv * norm_ws[relv], 1e-12f);
      ev = expf(logit_ws[t] - mono2f(m_ws[rowv]));  // unnormalized softmax
      atomicAdd(&ssum[rowv], ev);
    }
    sRow[w][lane] = rowv; sCol[w][lane] = colv; sRel[w][lane] = relv;
    sS[w][lane] = sc;     sE[w][lane] = ev;
  }
  asm volatile("s_wait_dscnt 0" ::: "memory");  // wave-local LDS handoff

  // --- stage rows: B128 gathers, B128 LDS stores ---
  for (int e = 0; e < TILE; ++e) {
    const int c = sCol[w][e];
    const int r = sRel[w][e];
    float4 f = reinterpret_cast<const float4*>(out + (size_t)c * (2 * DIMS))[lane];
    float4 g = reinterpret_cast<const float4*>(rel_emb + (size_t)r * DIMS)[lane];
    *reinterpret_cast<float4*>(&sA[w][e][lane * 4]) = f;
    *reinterpret_cast<float4*>(&sB[w][e][lane * 4]) = g;
  }
  asm volatile("s_wait_dscnt 0" ::: "memory");

  // --- matrix pipe: acc[16x16] += A(16xK) * B(Kx16), K=128, diag = dots ---
  // ISA 16x4 f32 A layout: lane holds row M=lane&15; lanes<16 carry K0/K1,
  // lanes>=16 carry K2/K3 (B symmetric with its column as N).
  const int mrow = lane & 15;
  const int koff = (lane < 16) ? 0 : 2;
  const float* Ar = &sA[w][mrow][0];
  const float* Br = &sB[w][mrow][0];
  v8f acc = {0.f, 0.f, 0.f, 0.f, 0.f, 0.f, 0.f, 0.f};
#pragma unroll
  for (int k = 0; k < DIMS / 4; ++k) {
    v2f a, b;
    a[0] = Ar[4 * k + koff]; a[1] = Ar[4 * k + koff + 1];
    b[0] = Br[4 * k + koff]; b[1] = Br[4 * k + koff + 1];
    acc = __builtin_amdgcn_wmma_f32_16x16x4_f32(false, a, false, b,
                                                (short)0, acc, false, false);
  }
  // diagonal lives at (vgpr i, lane i) for i<8 and (vgpr i-8, lane i+16)
  {
    const int di = (lane < 8) ? lane : (lane - 24);  // valid when selected
    float dv = 0.0f;
#pragma unroll
    for (int i = 0; i < 8; ++i)
      if (di == i) dv = acc[i];
    if (lane < 8)        sDot[w][lane] = dv;
    else if (lane >= 24) sDot[w][lane - 16] = dv;
  }
  asm volatile("s_wait_dscnt 0" ::: "memory");

  // --- scatter: e * (f - 2*(s*dot)*s*rel_raw) into accumulator half ---
#pragma unroll 1
  for (int e = 0; e < TILE; ++e) {
    const float ev = sE[w][e];
    if (ev == 0.0f) continue;  // LDS value: uniform across wave
    const float s = sS[w][e];
    const float coef = 2.0f * s * s * sDot[w][e];
    float4 f = *reinterpret_cast<const float4*>(&sA[w][e][lane * 4]);
    float4 g = *reinterpret_cast<const float4*>(&sB[w][e][lane * 4]);
    float* dst = out + (size_t)sRow[w][e] * (2 * DIMS) + DIMS + lane * 4;
    atomicAdd(dst + 0, ev * (f.x - coef * g.x));
    atomicAdd(dst + 1, ev * (f.y - coef * g.y));
    atomicAdd(dst + 2, ev * (f.z - coef * g.z));
    atomicAdd(dst + 3, ev * (f.w - coef * g.w));
  }
}

// ---------------------------------------------------------------------------
// Kernel 5: out[:,128:] = tanh(acc / ssum)  (0 for isolated rows)
__global__ void k_finalize(float* __restrict__ out,
                           const float* __restrict__ ssum, int N) {
  const int wid = threadIdx.x >> 5, lane = threadIdx.x & 31;
  const int n = blockIdx.x * (blockDim.x >> 5) + wid;
  if (n >= N) return;
  const float ss = ssum[n];
  const float inv = (ss > 0.0f) ? (1.0f / ss) : 0.0f;
  float* o = out + (size_t)n * (2 * DIMS) + DIMS;
  float4 a = reinterpret_cast<float4*>(o)[lane];
  float4 t;
  t.x = tanhf(a.x * inv); t.y = tanhf(a.y * inv);
  t.z = tanhf(a.z * inv); t.w = tanhf(a.w * inv);
  reinterpret_cast<float4*>(o)[lane] = t;
}

// ---------------------------------------------------------------------------
extern "C" void kernel_launch(void* const* d_in, const int* in_sizes, int n_in,
                              void* d_out, int out_size, void* d_ws,
                              size_t ws_size, hipStream_t stream) {
  const float* features = (const float*)d_in[0];
  const float* rel_emb  = (const float*)d_in[1];
  const float* r_val    = (const float*)d_in[2];
  const float* attn_k   = (const float*)d_in[3];
  const int*   adj      = (const int*)d_in[4];   // [2,T] row-major
  const int*   r_index  = (const int*)d_in[5];   // [2,T]; [1,:] = relation id
  float* out = (float*)d_out;                    // [N, 256]

  const int N = in_sizes[0] / DIMS;
  const int R = in_sizes[1] / DIMS;
  const int T = in_sizes[2];

  // workspace layout (~3.4 MB): logits[T] | rowmax[N] | rowsum[N] | norm[R] | adot[R]
  float*    logit_ws = (float*)d_ws;
  unsigned* m_ws     = (unsigned*)(logit_ws + T);
  float*    ssum     = (float*)(m_ws + N);
  float*    norm_ws  = ssum + N;
  float*    adot_ws  = norm_ws + R;

  k_nodes_init<<<(N + 7) / 8, 256, 0, stream>>>(features, out, m_ws, ssum, N);
  k_rel<<<(R + 7) / 8, 256, 0, stream>>>(rel_emb, attn_k, norm_ws, adot_ws, R);
  k_edge_logit<<<(T + 255) / 256, 256, 0, stream>>>(adj, r_index, r_val,
                                                    norm_ws, adot_ws,
                                                    logit_ws, m_ws, T);
  const int tiles = (T + TILE - 1) / TILE;
  k_edge_agg<<<(tiles + WV - 1) / WV, WV * 32, 0, stream>>>(
      rel_emb, adj, r_index, r_val, norm_ws, logit_ws, m_ws, ssum, out, T);
  k_finalize<<<(N + 7) / 8, 256, 0, stream>>>(out, ssum, N);
}